// SemGCN_Heatmaps_17171279249487
// MI455X (gfx1250) — compile-verified
//
#include <hip/hip_runtime.h>
#include <math.h>

typedef __attribute__((ext_vector_type(2))) float v2f;
typedef __attribute__((ext_vector_type(8))) float v8f;

#define ACT_NONE 0
#define ACT_RELU 1
#define ACT_SIGM 2

// Graph topology (EDGES start/end, EV/VE edge gathers), baked in.
__constant__ int d_es[12] = {0,1,2,1,4,0,6,7,8,7,10,9};
__constant__ int d_et[12] = {1,2,3,4,5,6,7,8,9,10,11,11};
__constant__ int d_ev[12] = {1,2,4,3,5,6,7,7,8,10,9,11};
__constant__ int d_ve[12] = {0,0,2,1,3,0,1,6,6,8,7,9};

static inline int cdiv(int a, int b) { return (a + b - 1) / b; }

// ---------------------------------------------------------------------------
// WMMA fp32 GEMM: Y(R x N) = act(X(R x K) @ W(K x N) + bias)
// One wave per 16x16 tile, K stepped by 4 via V_WMMA_F32_16X16X4_F32.
// Requires R%16==0, K%4==0, N%16==0.
// ---------------------------------------------------------------------------
__global__ __launch_bounds__(256) void gemm_wmma_kernel(
    const float* __restrict__ X, const float* __restrict__ W,
    const float* __restrict__ bias, float* __restrict__ Y,
    int R, int K, int N, int act) {
  int wavesPerBlock = blockDim.x >> 5;
  int waveId = blockIdx.x * wavesPerBlock + (threadIdx.x >> 5);
  int ntN = N >> 4;
  int ntR = R >> 4;
  if (waveId >= ntN * ntR) return;  // wave-uniform: EXEC stays all-ones inside
  int lane = threadIdx.x & 31;
  int rt = waveId / ntN, ct = waveId % ntN;
  int row0 = rt << 4, col0 = ct << 4;
  int m = lane & 15;
  int half = lane >> 4;  // 0: K+0/K+1 ; 1: K+2/K+3

  v8f acc = {0.f, 0.f, 0.f, 0.f, 0.f, 0.f, 0.f, 0.f};
  const float* xrow = X + (size_t)(row0 + m) * K;
  const float* wcol = W + col0 + m;
  for (int k = 0; k < K; k += 4) {
    v2f a, b;
    int k2 = k + 2 * half;
    a.x = xrow[k2];
    a.y = xrow[k2 + 1];
    b.x = wcol[(size_t)k2 * N];
    b.y = wcol[(size_t)(k2 + 1) * N];
#if __has_builtin(__builtin_amdgcn_wmma_f32_16x16x4_f32)
    acc = __builtin_amdgcn_wmma_f32_16x16x4_f32(
        /*neg_a=*/false, a, /*neg_b=*/false, b,
        /*c_mod=*/(short)0, acc, /*reuse_a=*/false, /*reuse_b=*/false);
#else
    // Fallback so the file still compiles if the builtin is absent
    acc[0] += a.x * b.x + a.y * b.y;
#endif
  }
  int col = col0 + m;
  float bv = bias ? bias[col] : 0.0f;
#pragma unroll
  for (int r = 0; r < 8; ++r) {
    float v = acc[r] + bv;
    if (act == ACT_RELU) v = v > 0.f ? v : 0.f;
    Y[(size_t)(row0 + r + 8 * half) * N + col] = v;
  }
}

// Scalar GEMM fallback for tiny shapes (K=2 inputs, N=2/3 heads)
__global__ void gemm_scalar_kernel(const float* __restrict__ X,
                                   const float* __restrict__ W,
                                   const float* __restrict__ bias,
                                   float* __restrict__ Y,
                                   int R, int K, int N, int act) {
  int idx = blockIdx.x * blockDim.x + threadIdx.x;
  if (idx >= R * N) return;
  int r = idx / N, n = idx % N;
  float acc = bias ? bias[n] : 0.f;
  for (int k = 0; k < K; ++k) acc += X[(size_t)r * K + k] * W[(size_t)k * N + n];
  if (act == ACT_RELU) acc = acc > 0.f ? acc : 0.f;
  else if (act == ACT_SIGM) acc = 1.f / (1.f + expf(-acc));
  Y[idx] = acc;
}

// ---------------------------------------------------------------------------
// Small utility kernels
// ---------------------------------------------------------------------------
__global__ void softmax12_kernel(const float* __restrict__ e, float* __restrict__ A) {
  int n = threadIdx.x;
  if (n >= 12) return;
  float mx = -INFINITY;
  for (int m2 = 0; m2 < 12; ++m2) mx = fmaxf(mx, e[n * 12 + m2]);
  float ex[12], s = 0.f;
  for (int m2 = 0; m2 < 12; ++m2) { ex[m2] = expf(e[n * 12 + m2] - mx); s += ex[m2]; }
  for (int m2 = 0; m2 < 12; ++m2) A[n * 12 + m2] = ex[m2] / s;
}

// out[b,n,d] = A[n,n]*h0[b,n,d] + sum_{m!=n} A[n,m]*h1[b,m,d] + bias[d]
__global__ void sem_combine_kernel(const float* __restrict__ h0,
                                   const float* __restrict__ h1,
                                   const float* __restrict__ A,
                                   const float* __restrict__ bias,
                                   float* __restrict__ Y, int C) {
  int idx = blockIdx.x * blockDim.x + threadIdx.x;
  if (idx >= 192 * C) return;
  int r = idx / C, c = idx % C;
  int b = r / 12, n = r % 12;
  float acc = A[n * 12 + n] * h0[idx];
  for (int m2 = 0; m2 < 12; ++m2) {
    if (m2 == n) continue;
    acc += A[n * 12 + m2] * h1[(size_t)(b * 12 + m2) * C + c];
  }
  Y[idx] = acc + bias[c];
}

// Per-channel mean / rstd over the 192 (batch*node) samples
__global__ void colstats_kernel(const float* __restrict__ X, float* __restrict__ mu,
                                float* __restrict__ rstd, int C) {
  __shared__ float ss[256];
  __shared__ float sq[256];
  int c = blockIdx.x, t = threadIdx.x;
  float v = (t < 192) ? X[(size_t)t * C + c] : 0.f;
  ss[t] = v; sq[t] = v * v;
  __syncthreads();
  for (int s = 128; s > 0; s >>= 1) {
    if (t < s) { ss[t] += ss[t + s]; sq[t] += sq[t + s]; }
    __syncthreads();
  }
  if (t == 0) {
    float m = ss[0] / 192.f;
    float var = sq[0] / 192.f - m * m;
    mu[c] = m;
    rstd[c] = rsqrtf(var + 1e-5f);
  }
}

__global__ void normrelu_kernel(const float* __restrict__ X, const float* __restrict__ mu,
                                const float* __restrict__ rstd,
                                const float* __restrict__ gamma,
                                const float* __restrict__ beta,
                                float* __restrict__ Y, int C) {
  int idx = blockIdx.x * blockDim.x + threadIdx.x;
  if (idx >= 192 * C) return;
  int c = idx % C;
  float v = gamma[c] * (X[idx] - mu[c]) * rstd[c] + beta[c];
  Y[idx] = v > 0.f ? v : 0.f;
}

__global__ void add_kernel(const float* __restrict__ Xa, const float* __restrict__ Xb,
                           float* __restrict__ Y, int n) {
  int i = blockIdx.x * blockDim.x + threadIdx.x;
  if (i < n) Y[i] = Xa[i] + Xb[i];
}

// pooled[b,d] = mean over 12 nodes
__global__ void rowpool_kernel(const float* __restrict__ X, float* __restrict__ P, int C) {
  int idx = blockIdx.x * blockDim.x + threadIdx.x;
  if (idx >= 16 * C) return;
  int b = idx / C, d = idx % C;
  float s = 0.f;
  for (int n = 0; n < 12; ++n) s += X[(size_t)(b * 12 + n) * C + d];
  P[idx] = s * (1.f / 12.f);
}

__global__ void scale_kernel(float* __restrict__ X, const float* __restrict__ S, int C) {
  int idx = blockIdx.x * blockDim.x + threadIdx.x;
  if (idx >= 192 * C) return;
  int b = idx / (12 * C), d = idx % C;
  X[idx] *= S[(size_t)b * C + d];
}

// x2 = x[...,:2]; y = SUB @ x2 (edge differences)
__global__ void prep_kernel(const float* __restrict__ x, float* __restrict__ X2,
                            float* __restrict__ Y2) {
  int idx = blockIdx.x * blockDim.x + threadIdx.x;
  if (idx >= 384) return;
  int r = idx >> 1, c = idx & 1;
  int b = r / 12, n = r % 12;
  X2[idx] = x[(size_t)r * 3 + c];
  Y2[idx] = x[(size_t)(b * 12 + d_et[n]) * 3 + c] - x[(size_t)(b * 12 + d_es[n]) * 3 + c];
}

// cat = [gout[start], gout[end], eout] for each edge
__global__ void eagg_gather_kernel(const float* __restrict__ G, const float* __restrict__ Ein,
                                   float* __restrict__ cat, int Cg, int Ce) {
  int Ccat = 2 * Cg + Ce;
  int idx = blockIdx.x * blockDim.x + threadIdx.x;
  if (idx >= 192 * Ccat) return;
  int r = idx / Ccat, c = idx % Ccat;
  int b = r / 12, e = r % 12;
  float v;
  if (c < Cg) v = G[(size_t)(b * 12 + d_es[e]) * Cg + c];
  else if (c < 2 * Cg) v = G[(size_t)(b * 12 + d_et[e]) * Cg + (c - Cg)];
  else v = Ein[(size_t)(b * 12 + e) * Ce + (c - 2 * Cg)];
  cat[idx] = v;
}

// cat = [gout, eout[ev[j]], eout[ve[j]]] for each joint
__global__ void jagg_gather_kernel(const float* __restrict__ G, const float* __restrict__ Ein,
                                   float* __restrict__ cat, int Cg, int Ce) {
  int Ccat = Cg + 2 * Ce;
  int idx = blockIdx.x * blockDim.x + threadIdx.x;
  if (idx >= 192 * Ccat) return;
  int r = idx / Ccat, c = idx % Ccat;
  int b = r / 12, j = r % 12;
  float v;
  if (c < Cg) v = G[(size_t)r * Cg + c];
  else if (c < Cg + Ce) v = Ein[(size_t)(b * 12 + d_ev[j]) * Ce + (c - Cg)];
  else v = Ein[(size_t)(b * 12 + d_ve[j]) * Ce + (c - Cg - Ce)];
  cat[idx] = v;
}

__global__ void concat2_kernel(const float* __restrict__ X1, const float* __restrict__ X2,
                               float* __restrict__ Y, int C1, int C2) {
  int Ccat = C1 + C2;
  int idx = blockIdx.x * blockDim.x + threadIdx.x;
  if (idx >= 192 * Ccat) return;
  int r = idx / Ccat, c = idx % Ccat;
  Y[idx] = (c < C1) ? X1[(size_t)r * C1 + c] : X2[(size_t)r * C2 + (c - C1)];
}

// ---------------------------------------------------------------------------
// Heatmap feature extraction (nearest resize: src = i*step + off)
// ---------------------------------------------------------------------------
__global__ void hmmax_kernel(const float* __restrict__ hm, float* __restrict__ hmx,
                             int H, int W, int step, int off) {
  int b = blockIdx.x / 12, j = blockIdx.x % 12;
  int t = threadIdx.x;
  float mx = -INFINITY;
  for (int i = t; i < H * W; i += 256) {
    int h = i / W, w = i % W;
    mx = fmaxf(mx, hm[((size_t)(b * 17 + 5 + j) * 64 + (h * step + off)) * 48 + (w * step + off)]);
  }
  __shared__ float red[256];
  red[t] = mx; __syncthreads();
  for (int s = 128; s > 0; s >>= 1) {
    if (t < s) red[t] = fmaxf(red[t], red[t + s]);
    __syncthreads();
  }
  if (t == 0) hmx[blockIdx.x] = red[0];
}

// jf[b,j,c] = mean_hw(feat*hm/hmmax) + max_hw(feat*hm/hmmax)
__global__ __launch_bounds__(256) void extract_kernel(
    const float* __restrict__ feat, const float* __restrict__ hm,
    const float* __restrict__ hmx, float* __restrict__ jf,
    int C, int H, int W, int step, int off) {
  int b = blockIdx.x / C, c = blockIdx.x % C;
  int t = threadIdx.x;
  float sumj[12], maxj[12];
#pragma unroll
  for (int j = 0; j < 12; ++j) { sumj[j] = 0.f; maxj[j] = -INFINITY; }
  int HW = H * W;
  for (int i = t; i < HW; i += 256) {
    int h = i / W, w = i % W;
    float f = feat[((size_t)(b * C + c) * H + h) * W + w];
#pragma unroll
    for (int j = 0; j < 12; ++j) {
      float hv = hm[((size_t)(b * 17 + 5 + j) * 64 + (h * step + off)) * 48 + (w * step + off)];
      float p = f * hv;
      sumj[j] += p;
      maxj[j] = fmaxf(maxj[j], p);
    }
  }
  __shared__ float red[256];
  __shared__ float sS[12], sM[12];
  for (int j = 0; j < 12; ++j) {
    red[t] = sumj[j]; __syncthreads();
    for (int s = 128; s > 0; s >>= 1) { if (t < s) red[t] += red[t + s]; __syncthreads(); }
    if (t == 0) sS[j] = red[0];
    __syncthreads();
    red[t] = maxj[j]; __syncthreads();
    for (int s = 128; s > 0; s >>= 1) { if (t < s) red[t] = fmaxf(red[t], red[t + s]); __syncthreads(); }
    if (t == 0) sM[j] = red[0];
    __syncthreads();
  }
  if (t < 12) {
    float mx = hmx[b * 12 + t];
    jf[(size_t)(b * 12 + t) * C + c] = sS[t] / ((float)HW * mx) + sM[t] / mx;
  }
}

// ---------------------------------------------------------------------------
// Host-side orchestration
// ---------------------------------------------------------------------------
struct GcP  { const float *W, *b, *beta, *e, *gamma; };
struct ResP { const float *b1, *b2, *w1, *w2; GcP g1, g2; };
struct SemP { const float *W, *b, *e; };
struct AggP { const float *W, *b; };

struct Ctx {
  float *A, *A2, *Bb, *D, *C1, *E, *F;
  float *T1, *T2, *T3, *H0, *H1, *MU, *RS, *POOL, *S1, *SB, *AB;
  hipStream_t st;
};

static void launch_gemm(const float* X, const float* W, const float* bias, float* Y,
                        int R, int K, int N, int act, hipStream_t st) {
  if ((K % 4) == 0 && (N % 16) == 0 && (R % 16) == 0 && K >= 4) {
    int tiles = (R / 16) * (N / 16);
    gemm_wmma_kernel<<<cdiv(tiles, 8), 256, 0, st>>>(X, W, bias, Y, R, K, N, act);
  } else {
    gemm_scalar_kernel<<<cdiv(R * N, 256), 256, 0, st>>>(X, W, bias, Y, R, K, N, act);
  }
}

// _sem: softmax adjacency, two projections, diag/off-diag combine (+bias)
static void run_sem(const float* X, const float* W, const float* bias, const float* e,
                    float* OUT, int K, int C, Ctx& x) {
  softmax12_kernel<<<1, 16, 0, x.st>>>(e, x.AB);
  launch_gemm(X, W, nullptr, x.H0, 192, K, C, ACT_NONE, x.st);
  launch_gemm(X, W + (size_t)K * C, nullptr, x.H1, 192, K, C, ACT_NONE, x.st);
  sem_combine_kernel<<<cdiv(192 * C, 256), 256, 0, x.st>>>(x.H0, x.H1, x.AB, bias, OUT, C);
}

// _gc: _sem -> per-channel norm -> relu
static void run_gc(const float* X, const GcP& g, float* OUT, int K, int C, Ctx& x) {
  run_sem(X, g.W, g.b, g.e, x.T1, K, C, x);
  colstats_kernel<<<C, 256, 0, x.st>>>(x.T1, x.MU, x.RS, C);
  normrelu_kernel<<<cdiv(192 * C, 256), 256, 0, x.st>>>(x.T1, x.MU, x.RS, g.gamma, g.beta, OUT, C);
}

// _res: two gc blocks + residual + SE gating. OUT must differ from X.
static void run_res(const float* X, const ResP& r, float* OUT, int C, Ctx& x) {
  run_gc(X, r.g1, x.T2, C, C, x);
  run_gc(x.T2, r.g2, x.T3, C, C, x);
  add_kernel<<<cdiv(192 * C, 256), 256, 0, x.st>>>(X, x.T3, OUT, 192 * C);
  rowpool_kernel<<<cdiv(16 * C, 256), 256, 0, x.st>>>(OUT, x.POOL, C);
  int Ch = C / 4;
  gemm_scalar_kernel<<<cdiv(16 * Ch, 256), 256, 0, x.st>>>(x.POOL, r.w1, r.b1, x.S1, 16, C, Ch, ACT_RELU);
  gemm_scalar_kernel<<<cdiv(16 * C, 256), 256, 0, x.st>>>(x.S1, r.w2, r.b2, x.SB, 16, Ch, C, ACT_SIGM);
  scale_kernel<<<cdiv(192 * C, 256), 256, 0, x.st>>>(OUT, x.SB, C);
}

extern "C" void kernel_launch(void* const* d_in, const int* in_sizes, int n_in,
                              void* d_out, int out_size, void* d_ws, size_t ws_size,
                              hipStream_t stream) {
  if (n_in < 161) return;  // expect 5 inputs + 156 param leaves
  const float* x     = (const float*)d_in[0];
  const float* hm    = (const float*)d_in[1];
  const float* feat0 = (const float*)d_in[2];
  const float* feat1 = (const float*)d_in[3];
  const float* feat2 = (const float*)d_in[4];

  // --- params, alphabetical pytree-flatten order ---
  int idx = 5;
  const float* leafs[200];
  for (int i = 5; i < n_in && i < 200 + 5; ++i) leafs[i - 5] = (const float*)d_in[i];
  int li = 0;
  auto NEXT = [&]() { return leafs[li++]; };
  // (lambda only reads pointers; kernels launched from plain statements below)
  AggP ae[4], aj[4];
  for (int i = 0; i < 4; ++i) { ae[i].W = NEXT(); ae[i].b = NEXT(); }
  for (int i = 0; i < 4; ++i) { aj[i].W = NEXT(); aj[i].b = NEXT(); }
  auto readGc = [&]() { GcP g; g.W = NEXT(); g.b = NEXT(); g.beta = NEXT(); g.e = NEXT(); g.gamma = NEXT(); return g; };
  auto readRes = [&]() { ResP r; r.b1 = NEXT(); r.b2 = NEXT(); r.g1 = readGc(); r.g2 = readGc(); r.w1 = NEXT(); r.w2 = NEXT(); return r; };
  auto readSem = [&]() { SemP s; s.W = NEXT(); s.b = NEXT(); s.e = NEXT(); return s; };
  ResP e1 = readRes(), e2 = readRes(), e3 = readRes(), e4 = readRes();
  GcP  ein = readGc();
  ResP g1 = readRes(), g2 = readRes(), g3 = readRes(), g4 = readRes();
  GcP  gin = readGc();
  SemP o1e = readSem(), o1g = readSem(), o2e = readSem(), o2g = readSem(),
       o3e = readSem(), o3g = readSem();
  (void)idx; (void)in_sizes; (void)out_size;

  // --- workspace bump allocator ---
  float* p = (float*)d_ws;
  size_t need = 0;
  auto alloc = [&](size_t n) { float* r = p; p += n; need += n; return r; };
  float* X2  = alloc(384);
  float* Y2  = alloc(384);
  float* jf0 = alloc(192 * 256);
  float* jf1 = alloc(192 * 128);
  float* jf2 = alloc(192 * 128);
  float* hmx0 = alloc(192);
  float* hmx1 = alloc(192);
  float* hmx2 = alloc(192);
  Ctx c;
  c.A  = alloc(192 * 640); c.A2 = alloc(192 * 640);
  c.Bb = alloc(192 * 128); c.D  = alloc(192 * 128);
  c.C1 = alloc(192 * 512);
  c.E  = alloc(192 * 1152); c.F = alloc(192 * 768);
  c.T1 = alloc(192 * 640); c.T2 = alloc(192 * 640); c.T3 = alloc(192 * 640);
  c.H0 = alloc(192 * 640); c.H1 = alloc(192 * 640);
  c.MU = alloc(640); c.RS = alloc(640);
  c.POOL = alloc(16 * 640); c.S1 = alloc(16 * 160); c.SB = alloc(16 * 640);
  c.AB = alloc(144);
  c.st = stream;
  if (need * sizeof(float) > ws_size) return;

  float* out = (float*)d_out;

  // --- heatmap-pooled joint features ---
  hmmax_kernel<<<192, 256, 0, stream>>>(hm, hmx0, 64, 48, 1, 0);
  hmmax_kernel<<<192, 256, 0, stream>>>(hm, hmx1, 32, 24, 2, 1);
  hmmax_kernel<<<192, 256, 0, stream>>>(hm, hmx2, 16, 12, 4, 2);
  extract_kernel<<<16 * 256, 256, 0, stream>>>(feat0, hm, hmx0, jf0, 256, 64, 48, 1, 0);
  extract_kernel<<<16 * 128, 256, 0, stream>>>(feat1, hm, hmx1, jf1, 128, 32, 24, 2, 1);
  extract_kernel<<<16 * 128, 256, 0, stream>>>(feat2, hm, hmx2, jf2, 128, 16, 12, 4, 2);

  // --- input stage ---
  prep_kernel<<<2, 256, 0, stream>>>(x, X2, Y2);
  run_gc(X2, gin, c.A, 2, 128, c);   // gout
  run_gc(Y2, ein, c.Bb, 2, 128, c);  // eout

  // --- agg round 0 ---
  eagg_gather_kernel<<<cdiv(192 * 384, 256), 256, 0, stream>>>(c.A, c.Bb, c.E, 128, 128);
  launch_gemm(c.E, ae[0].W, ae[0].b, c.D, 192, 384, 128, ACT_RELU, stream);   // eout1
  jagg_gather_kernel<<<cdiv(192 * 384, 256), 256, 0, stream>>>(c.A, c.Bb, c.F, 128, 128);
  launch_gemm(c.F, aj[0].W, aj[0].b, c.C1, 192, 384, 128, ACT_RELU, stream);  // gout1

  run_res(c.C1, g1, c.A, 128, c);   // gout
  run_res(c.D, e1, c.Bb, 128, c);   // eout

  // --- agg round 1 ---
  eagg_gather_kernel<<<cdiv(192 * 384, 256), 256, 0, stream>>>(c.A, c.Bb, c.E, 128, 128);
  launch_gemm(c.E, ae[1].W, ae[1].b, c.D, 192, 384, 128, ACT_RELU, stream);
  jagg_gather_kernel<<<cdiv(192 * 384, 256), 256, 0, stream>>>(c.A, c.Bb, c.F, 128, 128);
  launch_gemm(c.F, aj[1].W, aj[1].b, c.C1, 192, 384, 128, ACT_RELU, stream);

  concat2_kernel<<<cdiv(192 * 384, 256), 256, 0, stream>>>(c.C1, jf0, c.A, 128, 256);
  run_res(c.A, g2, c.A2, 384, c);   // gout (384) in A2
  run_res(c.D, e2, c.Bb, 128, c);   // eout
  run_sem(c.A2, o1g.W, o1g.b, o1g.e, out + 0,    384, 2, c);
  run_sem(c.Bb, o1e.W, o1e.b, o1e.e, out + 384,  128, 2, c);

  // --- agg round 2 ---
  eagg_gather_kernel<<<cdiv(192 * 896, 256), 256, 0, stream>>>(c.A2, c.Bb, c.E, 384, 128);
  launch_gemm(c.E, ae[2].W, ae[2].b, c.D, 192, 896, 128, ACT_RELU, stream);
  jagg_gather_kernel<<<cdiv(192 * 640, 256), 256, 0, stream>>>(c.A2, c.Bb, c.F, 384, 128);
  launch_gemm(c.F, aj[2].W, aj[2].b, c.C1, 192, 640, 384, ACT_RELU, stream);

  concat2_kernel<<<cdiv(192 * 512, 256), 256, 0, stream>>>(c.C1, jf1, c.A, 384, 128);
  run_res(c.A, g3, c.A2, 512, c);   // gout (512)
  run_res(c.D, e3, c.Bb, 128, c);
  run_sem(c.A2, o2g.W, o2g.b, o2g.e, out + 768,  512, 2, c);
  run_sem(c.Bb, o2e.W, o2e.b, o2e.e, out + 1152, 128, 2, c);

  // --- agg round 3 ---
  eagg_gather_kernel<<<cdiv(192 * 1152, 256), 256, 0, stream>>>(c.A2, c.Bb, c.E, 512, 128);
  launch_gemm(c.E, ae[3].W, ae[3].b, c.D, 192, 1152, 128, ACT_RELU, stream);
  jagg_gather_kernel<<<cdiv(192 * 768, 256), 256, 0, stream>>>(c.A2, c.Bb, c.F, 512, 128);
  launch_gemm(c.F, aj[3].W, aj[3].b, c.C1, 192, 768, 512, ACT_RELU, stream);

  concat2_kernel<<<cdiv(192 * 640, 256), 256, 0, stream>>>(c.C1, jf2, c.A, 512, 128);
  run_res(c.A, g4, c.A2, 640, c);   // gout (640)
  run_res(c.D, e4, c.Bb, 128, c);
  run_sem(c.A2, o3g.W, o3g.b, o3g.e, out + 1536, 640, 3, c);
  run_sem(c.Bb, o3e.W, o3e.b, o3e.e, out + 2112, 128, 3, c);
}